// PINN_75479755260340
// MI455X (gfx1250) — compile-verified
//
#include <hip/hip_runtime.h>
#include <hip/hip_bf16.h>

typedef float v2f __attribute__((ext_vector_type(2)));
typedef float v8f __attribute__((ext_vector_type(8)));

#define HDIM 256
#define HP   260                 // padded LDS row stride (floats) to avoid bank conflicts
#define PTS  32                  // points per workgroup
#define MS   (PTS * HP)          // LDS matrix stride (floats)
#define NQ   1000                // quadrature nodes

// ---------------------------------------------------------------------------
// Kernel 0: Gauss-Legendre nodes/weights (device-side replacement for leggauss)
// ---------------------------------------------------------------------------
__global__ __launch_bounds__(256) void PINN_gl_nodes_kernel(float* __restrict__ ext_x,
                                                            float* __restrict__ qw) {
    const int i = blockIdx.x * blockDim.x + threadIdx.x;
    const double PI = 3.14159265358979323846;
    if (i < NQ) {
        double x = cos(PI * ((double)i + 0.75) / ((double)NQ + 0.5));
        double dp = 1.0;
        for (int it = 0; it < 64; ++it) {
            double p0 = 1.0, p1 = x;
            for (int k = 2; k <= NQ; ++k) {
                double p2 = ((2.0 * k - 1.0) * x * p1 - (k - 1.0) * p0) / (double)k;
                p0 = p1; p1 = p2;
            }
            dp = (double)NQ * (x * p1 - p0) / (x * x - 1.0);
            double dx = p1 / dp;
            x -= dx;
            if (fabs(dx) < 1e-15) break;
        }
        double w = 2.0 / ((1.0 - x * x) * dp * dp);
        ext_x[i] = (float)(0.5 * x + 0.5);   // map [-1,1] -> [0, L], L = 1
        qw[i]    = (float)w;
    }
    if (i == NQ)     ext_x[NQ]     = 0.0f;   // boundary point x = 0
    if (i == NQ + 1) ext_x[NQ + 1] = 1.0f;   // boundary point x = L
}

// ---------------------------------------------------------------------------
// WMMA hidden layer: (h,h',h'')[32x256] @ W[256x256], tanh chain, into out LDS
// ---------------------------------------------------------------------------
__device__ __forceinline__ void layer_wmma(const float* __restrict__ in,
                                           float* __restrict__ outb,
                                           const float* __restrict__ Wt,
                                           const float* __restrict__ bias) {
    const int t    = threadIdx.x;
    const int wave = t >> 5;
    const int lane = t & 31;
    const int lhal = lane >> 4;       // 0: lanes 0-15, 1: lanes 16-31
    const int l15  = lane & 15;

    for (int r = 0; r < 2; ++r) {                  // row tile (points r*16 .. r*16+15)
        const float* arow = in + (r * 16 + l15) * HP;
        for (int nt = 0; nt < 2; ++nt) {           // this wave's two 16-col tiles
            const int n0 = wave * 32 + nt * 16;
            v8f acc_u = {}; v8f acc_p = {}; v8f acc_q = {};
            for (int k0 = 0; k0 < HDIM; k0 += 4) {
                const int ka = k0 + lhal * 2;      // A layout: V0 -> K, K+2 ; V1 -> K+1, K+3
                v2f a_u = *(const v2f*)(arow + 0 * MS + ka);
                v2f a_p = *(const v2f*)(arow + 1 * MS + ka);
                v2f a_q = *(const v2f*)(arow + 2 * MS + ka);
                v2f bf;                            // B layout mirrors A: V0 -> K, K+2
                bf.x = Wt[(size_t)(k0 + lhal * 2)     * HDIM + n0 + l15];
                bf.y = Wt[(size_t)(k0 + lhal * 2 + 1) * HDIM + n0 + l15];
                if (k0 + 8 < HDIM)                 // global_prefetch next K-panel
                    __builtin_prefetch(&Wt[(size_t)(k0 + 8 + lhal * 2) * HDIM + n0 + l15], 0, 3);
                acc_u = __builtin_amdgcn_wmma_f32_16x16x4_f32(false, a_u, false, bf,
                                                              (short)0, acc_u, false, false);
                acc_p = __builtin_amdgcn_wmma_f32_16x16x4_f32(false, a_p, false, bf,
                                                              (short)0, acc_p, false, false);
                acc_q = __builtin_amdgcn_wmma_f32_16x16x4_f32(false, a_q, false, bf,
                                                              (short)0, acc_q, false, false);
            }
            // bias + tanh hyper-dual chain, scatter to out buffer (C layout: VGPR v -> rows v, v+8)
            const int c = n0 + l15;
            const float bb = bias[c];
#pragma unroll
            for (int v = 0; v < 8; ++v) {
                const int p  = r * 16 + v + lhal * 8;
                const float z  = acc_u[v] + bb;
                const float zp = acc_p[v];
                const float zq = acc_q[v];
                const float h  = tanhf(z);
                const float om = 1.0f - h * h;
                const float hp = om * zp;
                const float hq = om * zq - 2.0f * h * hp * zp;
                outb[0 * MS + p * HP + c] = h;
                outb[1 * MS + p * HP + c] = hp;
                outb[2 * MS + p * HP + c] = hq;
            }
        }
    }
}

// ---------------------------------------------------------------------------
// Kernel 1: evaluate f, f', f'' at `count` points (32 points per workgroup)
// ---------------------------------------------------------------------------
__global__ __launch_bounds__(256) void PINN_eval_kernel(
    const float* __restrict__ xin, int count,
    const float* __restrict__ W1, const float* __restrict__ b1,
    const float* __restrict__ W2, const float* __restrict__ b2,
    const float* __restrict__ W3, const float* __restrict__ b3,
    const float* __restrict__ W4, const float* __restrict__ b4,
    float* __restrict__ out_u, float* __restrict__ out_du, float* __restrict__ out_ddu) {
    extern __shared__ float smem[];
    float* bufA = smem;            // 3 * 32 * HP floats
    float* bufB = smem + 3 * MS;
    __shared__ float s_pts[PTS];

    const int t    = threadIdx.x;
    const int base = blockIdx.x * PTS;

    if (t < PTS) {
        const int g = base + t;
        s_pts[t] = (g < count) ? xin[g] : 0.0f;
    }
    __syncthreads();

    // ---- Layer 1 (rank-1): z = s*W1 + b1, z' = W1, z'' = 0 ----
    {
        const int c = t;                    // one column per thread
        const float w = W1[c], bb = b1[c];
#pragma unroll 4
        for (int p = 0; p < PTS; ++p) {
            const float z  = s_pts[p] * w + bb;
            const float h  = tanhf(z);
            const float hp = (1.0f - h * h) * w;
            const float hq = -2.0f * h * hp * w;
            bufA[0 * MS + p * HP + c] = h;
            bufA[1 * MS + p * HP + c] = hp;
            bufA[2 * MS + p * HP + c] = hq;
        }
    }
    __syncthreads();

    layer_wmma(bufA, bufB, W2, b2);   // layer 2
    __syncthreads();
    layer_wmma(bufB, bufA, W3, b3);   // layer 3
    __syncthreads();

    // ---- Layer 4 (dot with W4): 8 threads per point ----
    {
        const int p = t >> 3;
        const int q = t & 7;
        float su = 0.0f, sp = 0.0f, sq = 0.0f;
#pragma unroll 4
        for (int c = q; c < HDIM; c += 8) {
            const float w = W4[c];
            su += bufA[0 * MS + p * HP + c] * w;
            sp += bufA[1 * MS + p * HP + c] * w;
            sq += bufA[2 * MS + p * HP + c] * w;
        }
        for (int off = 4; off; off >>= 1) {
            su += __shfl_down(su, off, 8);
            sp += __shfl_down(sp, off, 8);
            sq += __shfl_down(sq, off, 8);
        }
        if (q == 0) {
            const int g = base + p;
            if (g < count) {
                out_u[g]   = su + b4[0];
                out_du[g]  = sp;
                out_ddu[g] = sq;
            }
        }
    }
}

// ---------------------------------------------------------------------------
// Kernel 2: quadrature reduction + boundary terms -> total_loss
// ---------------------------------------------------------------------------
__global__ __launch_bounds__(256) void PINN_loss_kernel(const float* __restrict__ qw,
                                                        const float* __restrict__ ws_u,
                                                        const float* __restrict__ ws_du,
                                                        float* __restrict__ out_loss) {
    __shared__ float red[256];
    float s = 0.0f;
    for (int i = threadIdx.x; i < NQ; i += 256) {
        const float d = ws_du[i];
        s += qw[i] * d * d;
    }
    red[threadIdx.x] = s;
    __syncthreads();
    for (int off = 128; off; off >>= 1) {
        if (threadIdx.x < off) red[threadIdx.x] += red[threadIdx.x + off];
        __syncthreads();
    }
    if (threadIdx.x == 0) {
        const float EA   = 5000.0f * 0.0004f;          // E_MOD * AREA
        const float inter = 0.5f * EA * 0.5f * 1.0f * red[0];  // 0.5*EA*0.5*(b-a)*sum
        const float u0 = ws_u[NQ];
        const float uL = ws_u[NQ + 1];
        out_loss[0] = 1.0f * (inter - 5.0f * uL) + 1.0f * u0 * u0;  // W_PEL, F_LOAD, W_DIR
    }
}

// ---------------------------------------------------------------------------
extern "C" void kernel_launch(void* const* d_in, const int* in_sizes, int n_in,
                              void* d_out, int out_size, void* d_ws, size_t ws_size,
                              hipStream_t stream) {
    (void)n_in; (void)out_size; (void)ws_size;
    const float* x  = (const float*)d_in[0];
    const float* W1 = (const float*)d_in[1];
    const float* b1 = (const float*)d_in[2];
    const float* W2 = (const float*)d_in[3];
    const float* b2 = (const float*)d_in[4];
    const float* W3 = (const float*)d_in[5];
    const float* b3 = (const float*)d_in[6];
    const float* W4 = (const float*)d_in[7];
    const float* b4 = (const float*)d_in[8];
    float* out = (float*)d_out;
    float* ws  = (float*)d_ws;

    const int N = in_sizes[0];               // 131072

    float* ext_x  = ws;                      // [0, 1024): tn nodes + {0, L}
    float* qw     = ws + 1024;               // [1024, 2048)
    float* ws_u   = ws + 2048;               // eval at ext points
    float* ws_du  = ws + 3072;
    float* ws_ddu = ws + 4096;

    const size_t shmem = (size_t)2 * 3 * MS * sizeof(float);  // two ping-pong triples

    PINN_gl_nodes_kernel<<<4, 256, 0, stream>>>(ext_x, qw);

    PINN_eval_kernel<<<(N + PTS - 1) / PTS, 256, shmem, stream>>>(
        x, N, W1, b1, W2, b2, W3, b3, W4, b4, out, out + N, out + 2 * N);

    PINN_eval_kernel<<<(NQ + 2 + PTS - 1) / PTS, 256, shmem, stream>>>(
        ext_x, NQ + 2, W1, b1, W2, b2, W3, b3, W4, b4, ws_u, ws_du, ws_ddu);

    PINN_loss_kernel<<<1, 256, 0, stream>>>(qw, ws_u, ws_du, out + 3 * (size_t)N);
}